// HGraphSAGE_37838661877859
// MI455X (gfx1250) — compile-verified
//
#include <hip/hip_runtime.h>

typedef __attribute__((ext_vector_type(2))) float v2f;
typedef __attribute__((ext_vector_type(8))) float v8f;

static constexpr int HD = 64;   // hidden dim
static constexpr int NF = 200000, NA = 50000, NE = 400000, NL = 2;

// --- force global (addrspace 1) memory ops: decouple LOADcnt from DScnt ----
__device__ __forceinline__ v2f gload_v2f(const float* p) {
    return *(const __attribute__((address_space(1))) v2f*)(unsigned long long)p;
}
__device__ __forceinline__ float gload_f32(const float* p) {
    return *(const __attribute__((address_space(1))) float*)(unsigned long long)p;
}
__device__ __forceinline__ void gstore_f32(float* p, float v) {
    *(__attribute__((address_space(1))) float*)(unsigned long long)p = v;
}

// ---------------------------------------------------------------------------
// Fused GEMM: out[n, 0:64] = act( sum_j A_j[n, 0:K] @ W_j[0:64, 0:K]^T + b0 + b1 )
// One wave computes a 16x64 output tile via V_WMMA_F32_16X16X4_F32 (full f32).
// A-operand layout (16x4 f32): lane&15 = M row, lane>>4 selects K pair {0,1}/{2,3}.
// B-operand layout (4x16 f32): lane&15 = N col, lane>>4 selects K pair (B = W^T).
// D layout (16x16 f32): VGPR v -> row v (lanes 0-15) / row v+8 (lanes 16-31).
// ---------------------------------------------------------------------------
__global__ __launch_bounds__(256)
void sage_gemm_wmma(const float* __restrict__ A0, const float* __restrict__ W0,
                    const float* __restrict__ A1, const float* __restrict__ W1,
                    const float* __restrict__ A2, const float* __restrict__ W2,
                    const float* __restrict__ b0, const float* __restrict__ b1,
                    float* __restrict__ out, int nrows, int K, int npairs,
                    int do_relu)
{
    __shared__ float sW[3 * 64 * 64];   // 48 KB max (npairs*64*K floats used)

    const float* Ws[3] = {W0, W1, W2};
    const int tot = npairs * 64 * K;
    for (int i = threadIdx.x; i < tot; i += 256) {
        int j = i / (64 * K);
        sW[i] = gload_f32(Ws[j] + (i - j * 64 * K));
    }
    __syncthreads();

    const int wave = threadIdx.x >> 5;
    const int lane = threadIdx.x & 31;
    const int tile = blockIdx.x * 8 + wave;
    if (tile * 16 >= nrows) return;          // wave-uniform: EXEC stays all-ones

    const int rowBase = tile * 16;
    const int m    = lane & 15;              // M row (A) / N col (B)
    const int koff = (lane >> 4) << 1;       // 0 or 2: which K pair this half-wave holds

    v8f acc[4];
    const v8f vzero = {0.f, 0.f, 0.f, 0.f, 0.f, 0.f, 0.f, 0.f};
    acc[0] = vzero; acc[1] = vzero; acc[2] = vzero; acc[3] = vzero;

    const float* As[3] = {A0, A1, A2};
    for (int j = 0; j < npairs; ++j) {
        const float* A  = As[j] + (long long)rowBase * K + m * K + koff;
        const float* Wj = &sW[j * 64 * K];
        for (int kk = 0; kk < K; kk += 4) {
            const int kb = kk + koff;                       // even -> 8B aligned
            v2f a = gload_v2f(A + kk);                      // global_load_b64
#pragma unroll
            for (int nt = 0; nt < 4; ++nt) {
                v2f b = *(const v2f*)(Wj + (nt * 16 + m) * K + kb);  // ds_load
                acc[nt] = __builtin_amdgcn_wmma_f32_16x16x4_f32(
                    false, a, false, b, (short)0, acc[nt], false, false);
            }
        }
    }

    const int mo = (lane >> 4) << 3;         // lanes 16-31 hold rows M+8
#pragma unroll
    for (int nt = 0; nt < 4; ++nt) {
        const int n = nt * 16 + m;
        float bias = (b0 ? gload_f32(b0 + n) : 0.0f) + (b1 ? gload_f32(b1 + n) : 0.0f);
#pragma unroll
        for (int v = 0; v < 8; ++v) {
            float x = acc[nt][v] + bias;
            if (do_relu) x = fmaxf(x, 0.0f);
            gstore_f32(out + (long long)(rowBase + mo + v) * HD + n, x);
        }
    }
}

// ---------------------------------------------------------------------------
__global__ void zero_f32_kernel(float* __restrict__ p, int n) {
    int i = blockIdx.x * blockDim.x + threadIdx.x;
    if (i < n) gstore_f32(p + i, 0.0f);
}

__global__ void mat_add_kernel(const float* __restrict__ a, const float* __restrict__ b,
                               float* __restrict__ c, int n) {
    int i = blockIdx.x * blockDim.x + threadIdx.x;
    if (i < n) gstore_f32(c + i, gload_f32(a + i) + gload_f32(b + i));
}

// one thread per (edge, feature): agg[dst[e], f] += h[src[e], f]
__global__ void scatter_feat_kernel(const float* __restrict__ h,
                                    const int* __restrict__ src,
                                    const int* __restrict__ dst,
                                    float* __restrict__ agg, int E) {
    int gid = blockIdx.x * blockDim.x + threadIdx.x;
    if (gid >= E * HD) return;
    int e = gid >> 6, f = gid & 63;
    float v = gload_f32(h + src[e] * HD + f);
    atomicAdd(&agg[dst[e] * HD + f], v);
}

__global__ void scatter_cnt_kernel(const int* __restrict__ dst,
                                   float* __restrict__ cnt, int E) {
    int e = blockIdx.x * blockDim.x + threadIdx.x;
    if (e < E) atomicAdd(&cnt[dst[e]], 1.0f);
}

__global__ void normalize_kernel(float* __restrict__ agg,
                                 const float* __restrict__ cnt, int nrows) {
    int gid = blockIdx.x * blockDim.x + threadIdx.x;
    if (gid >= nrows * HD) return;
    gstore_f32(agg + gid, gload_f32(agg + gid) / fmaxf(gload_f32(cnt + (gid >> 6)), 1.0f));
}

// column sums for mean pool: blockDim = 64 (one feature per thread)
__global__ void colsum_kernel(const float* __restrict__ h, int nrows,
                              float* __restrict__ outsum) {
    const int f = threadIdx.x;
    long long r0 = (long long)blockIdx.x * 2048;
    long long r1 = r0 + 2048; if (r1 > nrows) r1 = nrows;
    float s = 0.0f;
    for (long long r = r0; r < r1; ++r) s += gload_f32(h + r * HD + f);
    atomicAdd(&outsum[f], s);
}

__global__ void classifier_kernel(const float* __restrict__ poolF,
                                  const float* __restrict__ poolA,
                                  float nF, float nA,
                                  const float* __restrict__ Wc1, const float* __restrict__ bc1,
                                  const float* __restrict__ Wc2, const float* __restrict__ bc2,
                                  float* __restrict__ out) {
    __shared__ float pooled[128];
    __shared__ float hidden[64];
    const int t = threadIdx.x;  // 128 threads
    pooled[t] = (t < 64) ? (gload_f32(poolF + t) / nF) : (gload_f32(poolA + t - 64) / nA);
    __syncthreads();
    if (t < 64) {
        float s = gload_f32(bc1 + t);
        for (int k = 0; k < 128; ++k) s += pooled[k] * gload_f32(Wc1 + t * 128 + k);
        hidden[t] = fmaxf(s, 0.0f);
    }
    __syncthreads();
    if (t < 2) {
        float s = gload_f32(bc2 + t);
        for (int k = 0; k < 64; ++k) s += hidden[k] * gload_f32(Wc2 + t * 64 + k);
        gstore_f32(out + t, s);
    }
}

// ---------------------------------------------------------------------------
extern "C" void kernel_launch(void* const* d_in, const int* in_sizes, int n_in,
                              void* d_out, int out_size, void* d_ws, size_t ws_size,
                              hipStream_t stream) {
    const float* x_func    = (const float*)d_in[0];
    const float* x_api     = (const float*)d_in[1];
    const float* w_in_func = (const float*)d_in[2];
    const float* b_in_func = (const float*)d_in[3];
    const float* w_in_api  = (const float*)d_in[4];
    const float* b_in_api  = (const float*)d_in[5];
    const float* Wl_calls  = (const float*)d_in[6];
    const float* bl_calls  = (const float*)d_in[7];
    const float* Wr_calls  = (const float*)d_in[8];
    const float* Wl_uses   = (const float*)d_in[9];
    const float* bl_uses   = (const float*)d_in[10];
    const float* Wr_uses   = (const float*)d_in[11];
    const float* Wl_usedby = (const float*)d_in[12];
    const float* bl_usedby = (const float*)d_in[13];
    const float* Wr_usedby = (const float*)d_in[14];
    const float* Wc1       = (const float*)d_in[15];
    const float* bc1       = (const float*)d_in[16];
    const float* Wc2       = (const float*)d_in[17];
    const float* bc2       = (const float*)d_in[18];
    const int* ei_calls  = (const int*)d_in[19];
    const int* ei_uses   = (const int*)d_in[20];
    const int* ei_usedby = (const int*)d_in[21];

    // ---- workspace carve (256B aligned) ----
    char* p = (char*)d_ws;
    auto carve = [&](size_t bytes) -> float* {
        float* r = (float*)p;
        p += (bytes + 255) & ~(size_t)255;
        return r;
    };
    float* h_func  = carve((size_t)NF * HD * 4);
    float* h_func2 = carve((size_t)NF * HD * 4);
    float* agg_c   = carve((size_t)NF * HD * 4);
    float* agg_u   = carve((size_t)NF * HD * 4);
    float* h_api   = carve((size_t)NA * HD * 4);
    float* h_api2  = carve((size_t)NA * HD * 4);
    float* agg_a   = carve((size_t)NA * HD * 4);
    float* cnt_c   = carve((size_t)NF * 4);
    float* cnt_u   = carve((size_t)NF * 4);
    float* cnt_a   = carve((size_t)NA * 4);
    float* Wsum    = carve((size_t)64 * 64 * 4);
    float* pool    = carve((size_t)128 * 4);

    const int gemmBlocksF = (NF / 16 + 7) / 8;   // 8 waves (tiles) per block
    const int gemmBlocksA = (NA / 16 + 7) / 8;

    // ---- input projections (K=128 / K=64), relu ----
    sage_gemm_wmma<<<gemmBlocksF, 256, 0, stream>>>(
        x_func, w_in_func, nullptr, nullptr, nullptr, nullptr,
        b_in_func, nullptr, h_func, NF, 128, 1, 1);
    sage_gemm_wmma<<<gemmBlocksA, 256, 0, stream>>>(
        x_api, w_in_api, nullptr, nullptr, nullptr, nullptr,
        b_in_api, nullptr, h_api, NA, 64, 1, 1);

    const int efThreads = NE * HD;
    for (int l = 0; l < NL; ++l) {
        const float* WlC = Wl_calls  + l * 64 * 64;  const float* blC = bl_calls  + l * 64;
        const float* WrC = Wr_calls  + l * 64 * 64;
        const float* WlU = Wl_uses   + l * 64 * 64;  const float* blU = bl_uses   + l * 64;
        const float* WrU = Wr_uses   + l * 64 * 64;
        const float* WlB = Wl_usedby + l * 64 * 64;  const float* blB = bl_usedby + l * 64;
        const float* WrB = Wr_usedby + l * 64 * 64;

        // agg_c = scatter-mean(h_func) over calls (func -> func)
        zero_f32_kernel<<<(NF * HD + 255) / 256, 256, 0, stream>>>(agg_c, NF * HD);
        zero_f32_kernel<<<(NF + 255) / 256, 256, 0, stream>>>(cnt_c, NF);
        scatter_feat_kernel<<<(efThreads + 255) / 256, 256, 0, stream>>>(
            h_func, ei_calls, ei_calls + NE, agg_c, NE);
        scatter_cnt_kernel<<<(NE + 255) / 256, 256, 0, stream>>>(ei_calls + NE, cnt_c, NE);
        normalize_kernel<<<(NF * HD + 255) / 256, 256, 0, stream>>>(agg_c, cnt_c, NF);

        // agg_u = scatter-mean(h_api) over used_by (api -> func)
        zero_f32_kernel<<<(NF * HD + 255) / 256, 256, 0, stream>>>(agg_u, NF * HD);
        zero_f32_kernel<<<(NF + 255) / 256, 256, 0, stream>>>(cnt_u, NF);
        scatter_feat_kernel<<<(efThreads + 255) / 256, 256, 0, stream>>>(
            h_api, ei_usedby, ei_usedby + NE, agg_u, NE);
        scatter_cnt_kernel<<<(NE + 255) / 256, 256, 0, stream>>>(ei_usedby + NE, cnt_u, NE);
        normalize_kernel<<<(NF * HD + 255) / 256, 256, 0, stream>>>(agg_u, cnt_u, NF);

        // agg_a = scatter-mean(h_func) over uses (func -> api)
        zero_f32_kernel<<<(NA * HD + 255) / 256, 256, 0, stream>>>(agg_a, NA * HD);
        zero_f32_kernel<<<(NA + 255) / 256, 256, 0, stream>>>(cnt_a, NA);
        scatter_feat_kernel<<<(efThreads + 255) / 256, 256, 0, stream>>>(
            h_func, ei_uses, ei_uses + NE, agg_a, NE);
        scatter_cnt_kernel<<<(NE + 255) / 256, 256, 0, stream>>>(ei_uses + NE, cnt_a, NE);
        normalize_kernel<<<(NA * HD + 255) / 256, 256, 0, stream>>>(agg_a, cnt_a, NA);

        // Wsum = Wr_calls[l] + Wr_usedby[l]   (h_func hits both lin_r paths)
        mat_add_kernel<<<(64 * 64 + 255) / 256, 256, 0, stream>>>(WrC, WrB, Wsum, 64 * 64);

        // out_func = relu(agg_c@WlC^T + agg_u@WlB^T + h_func@Wsum^T + blC + blB)
        sage_gemm_wmma<<<gemmBlocksF, 256, 0, stream>>>(
            agg_c, WlC, agg_u, WlB, h_func, Wsum, blC, blB, h_func2, NF, 64, 3, 1);
        // out_api = relu(agg_a@WlU^T + h_api@WrU^T + blU)
        sage_gemm_wmma<<<gemmBlocksA, 256, 0, stream>>>(
            agg_a, WlU, h_api, WrU, nullptr, nullptr, blU, nullptr, h_api2, NA, 64, 2, 1);

        float* t;
        t = h_func; h_func = h_func2; h_func2 = t;
        t = h_api;  h_api  = h_api2;  h_api2  = t;
    }

    // ---- mean pool per type + classifier ----
    zero_f32_kernel<<<1, 128, 0, stream>>>(pool, 128);
    colsum_kernel<<<(NF + 2047) / 2048, 64, 0, stream>>>(h_func, NF, pool);
    colsum_kernel<<<(NA + 2047) / 2048, 64, 0, stream>>>(h_api, NA, pool + 64);
    classifier_kernel<<<1, 128, 0, stream>>>(pool, pool + 64, (float)NF, (float)NA,
                                             Wc1, bc1, Wc2, bc2, (float*)d_out);
}